// SampleDepthwise_89507118448688
// MI455X (gfx1250) — compile-verified
//
#include <hip/hip_runtime.h>

typedef __attribute__((ext_vector_type(2))) float v2f;
typedef __attribute__((ext_vector_type(8))) float v8f;

#define B_ 8
#define C_ 256
#define H_ 96
#define W_ 96
#define S_ 4
#define K_ 3
#define HW_ (H_ * W_)             // 9216
#define TILES_PER_ROW (W_ / 16)   // 6
#define PIX_TILES (B_ * H_ * TILES_PER_ROW)  // 4608

// One wave computes a 16-channel x 16-pixel output tile.
// Per tap: sampled(16x16) = Weight(16ch x 16scope) @ Coef(16scope x 16px) via
// 4 chained v_wmma_f32_16x16x4_f32. The scope-coef matrix is separable
// (coef[y*4+x] = fy[y]*fx[x]), so each lane builds its B fragments in
// registers from its pixel's rotation values -- no LDS, no scatter. The 0/1
// zero-padding mask is folded into the B column (bit-exact, since the per-tap
// accumulator starts at zero), then acc += sampled * patch, branch-free.
__global__ __launch_bounds__(256) void SampleDepthwise_wmma_kernel(
    const float* __restrict__ x,
    const float* __restrict__ rot,
    const float* __restrict__ weight,
    float* __restrict__ out)
{
  const int wave = threadIdx.x >> 5;
  const int lane = threadIdx.x & 31;
  const int half = lane >> 4;            // 0: lanes 0-15, 1: lanes 16-31
  const int pix  = lane & 15;

  // 8 waves of a block share one pixel tile, covering 8 channel groups
  const int pt  = blockIdx.x >> 1;                  // pixel tile id
  const int cg  = ((blockIdx.x & 1) << 3) + wave;   // channel group 0..15
  const int b   = pt / (H_ * TILES_PER_ROW);
  const int rem = pt % (H_ * TILES_PER_ROW);
  const int h   = rem / TILES_PER_ROW;
  const int w0  = (rem % TILES_PER_ROW) * 16;
  const int c0  = cg * 16;

  // A fragments: 16 channels x 16 scope weights, split into 4 K=4 fragments.
  // 16x4 fp32 A layout: lanes 0-15 hold K={0,1}, lanes 16-31 hold K={2,3}.
  v2f a[4];
  {
    const int ch_a = c0 + pix;
    #pragma unroll
    for (int j = 0; j < 4; ++j) {
      const int k = 4 * j + 2 * half;
      a[j].x = weight[ch_a * 16 + k];
      a[j].y = weight[ch_a * 16 + k + 1];
    }
  }

  // Hoisted per-lane base pointers; taps/channels become constant immediates.
  const float* rotp = rot + (size_t)b * 18 * HW_ + h * W_ + (w0 + pix);
  const float* xch  = x   + ((size_t)b * C_ + c0 + 8 * half) * HW_;
  float*       och  = out + ((size_t)b * C_ + c0 + 8 * half) * HW_;

  const float fx_lo = (float)(2 * half);      // x-position of this lane's B col pair
  const float fx_hi = (float)(2 * half + 1);

  v8f acc = {};

  #pragma unroll
  for (int kh = 0; kh < K_; ++kh) {
    #pragma unroll
    for (int kw = 0; kw < K_; ++kw) {
      const int tap = kh * K_ + kw;

      // Rotation for this lane's pixel (lanes 16-31 dup-load lanes 0-15's
      // addresses; L0 hit). Tap select = constant immediate offset.
      const float ry = rotp[(2 * tap + 0) * HW_];
      const float rx = rotp[(2 * tap + 1) * HW_];
      const float py = (kh + 0.5f) * ((float)S_ / (float)K_) - 0.5f + ry;
      const float px = (kw + 0.5f) * ((float)S_ / (float)K_) - 0.5f + rx;

      // Zero-padding mask for this tap/pixel, folded into the B column.
      const int hh    = h + kh - 1;
      const int ww    = w0 + pix + kw - 1;
      const int hh_cl = min(max(hh, 0), H_ - 1);
      const int ww_cl = min(max(ww, 0), W_ - 1);
      const float m   = ((hh == hh_cl) & (ww == ww_cl)) ? 1.0f : 0.0f;

      // Separable bilinear footprint over the 4x4 scope kernel:
      // fy[y] = max(0, 1-|py-y|), fx[x] = max(0, 1-|px-x|)  (then x-mask m).
      float fy[4];
      #pragma unroll
      for (int y = 0; y < 4; ++y)
        fy[y] = fmaxf(0.0f, 1.0f - fabsf(py - (float)y));
      const float fx0 = fmaxf(0.0f, 1.0f - fabsf(px - fx_lo)) * m;
      const float fx1 = fmaxf(0.0f, 1.0f - fabsf(px - fx_hi)) * m;

      // Chained WMMA over the 16 scope positions (fragment j = scope row y=j).
      // B 4x16 fp32 layout mirrors A: lanes 0-15 rows K={0,1}, 16-31 K={2,3}.
      v8f c = {};
      #pragma unroll
      for (int j = 0; j < 4; ++j) {
        v2f bb;
        bb.x = fy[j] * fx0;
        bb.y = fy[j] * fx1;
        c = __builtin_amdgcn_wmma_f32_16x16x4_f32(
            /*neg_a=*/false, a[j], /*neg_b=*/false, bb,
            /*c_mod=*/(short)0, c, /*reuse_a=*/false, /*reuse_b=*/false);
      }

      // acc += sampled * patch (clamped address is always valid; OOB pixels
      // already zeroed via the mask folded into B).
      const float* xp = xch + hh_cl * W_ + ww_cl;
      #pragma unroll
      for (int i = 0; i < 8; ++i) {
        acc[i] += c[i] * xp[i * HW_];
      }
    }
  }

  // Store the 16x16 tile (C/D layout: VGPR i = rows M=i / M=i+8, N = pixel).
  // Non-temporal: out is write-once; keep L2 for the x working set.
  float* op = och + h * W_ + (w0 + pix);
  #pragma unroll
  for (int i = 0; i < 8; ++i) {
    __builtin_nontemporal_store(acc[i], op + i * HW_);
  }
}

extern "C" void kernel_launch(void* const* d_in, const int* in_sizes, int n_in,
                              void* d_out, int out_size, void* d_ws, size_t ws_size,
                              hipStream_t stream) {
  (void)in_sizes; (void)out_size; (void)d_ws; (void)ws_size;
  if (n_in < 3) return;
  const float* x   = (const float*)d_in[0];
  const float* rot = (const float*)d_in[1];
  const float* w   = (const float*)d_in[2];
  float* out = (float*)d_out;

  const int blocks = PIX_TILES * 2;  // 9216 blocks x 256 threads (8 waves each)
  SampleDepthwise_wmma_kernel<<<blocks, 256, 0, stream>>>(x, rot, w, out);
}